// JumpingKnowledgeNetwork_40853728919686
// MI455X (gfx1250) — compile-verified
//
#include <hip/hip_runtime.h>
#include <hip/hip_bf16.h>

typedef __attribute__((ext_vector_type(2))) float v2f;
typedef __attribute__((ext_vector_type(8))) float v8f;

#define N_NODES 131072
#define NF      256
#define NH      256
#define N0S     4096
#define N1S     32768
#define N2S     131072
#define E1S     557056
#define E0S     69632
#define ALPHA   0.2f

// ---------------------------------------------------------------------------
// GEMM: C[M x 256] = gather(A)[M x 256] @ W[256 x 256], f32 via WMMA 16x16x4.
// Block = 256 threads (8 waves), block tile 64x64, wave tile 16x32.
// ---------------------------------------------------------------------------
#define KC   32
#define ASTR 36   // 64 rows x 32 cols, padded stride (conflict-free, float4-aligned)
#define BSTR 68   // 32 rows x 64 cols, padded stride

__global__ __launch_bounds__(256)
void gat_gemm_wmma(const float* __restrict__ A, const float* __restrict__ W,
                   const int* __restrict__ gather, float* __restrict__ C) {
    __shared__ float At[64 * ASTR];
    __shared__ float Bt[KC * BSTR];

    const int tid  = threadIdx.x;
    const int wave = tid >> 5;
    const int lane = tid & 31;
    const int wm = (wave & 3) * 16;   // wave row offset in block tile
    const int wn = (wave >> 2) * 32;  // wave col offset in block tile
    const long rowBase = (long)blockIdx.x * 64;
    const int  colBase = blockIdx.y * 64;

    v8f c0 = {0.f,0.f,0.f,0.f,0.f,0.f,0.f,0.f};
    v8f c1 = {0.f,0.f,0.f,0.f,0.f,0.f,0.f,0.f};

    const int khi   = (lane >> 4) * 2;     // 0 for lanes 0-15, 2 for 16-31
    const int mrow  = wm + (lane & 15);
    const int ncol0 = wn + (lane & 15);

    for (int k0 = 0; k0 < NF; k0 += KC) {
        // stage A tile 64x32 (with row gather): 512 float4, 2 per thread
        #pragma unroll
        for (int p = 0; p < 2; ++p) {
            int i4 = tid + p * 256;
            int r  = i4 >> 3;            // 8 float4 per row
            int c  = (i4 & 7) * 4;
            long gr  = rowBase + r;
            long src = gather ? (long)gather[gr] : gr;
            float4 v = *(const float4*)(A + src * NF + k0 + c);
            *(float4*)(At + r * ASTR + c) = v;
        }
        // stage B tile 32x64: 512 float4, 2 per thread
        #pragma unroll
        for (int p = 0; p < 2; ++p) {
            int i4 = tid + p * 256;
            int r  = i4 >> 4;            // 16 float4 per row
            int c  = (i4 & 15) * 4;
            float4 v = *(const float4*)(W + (long)(k0 + r) * NH + colBase + c);
            *(float4*)(Bt + r * BSTR + c) = v;
        }
        __syncthreads();

        #pragma unroll
        for (int kk = 0; kk < KC; kk += 4) {
            // A frag: VGPR0 = K(kk+khi), VGPR1 = K(kk+khi+1)   (ISA 7.12.2)
            v2f a;
            a.x = At[mrow * ASTR + kk + khi];
            a.y = At[mrow * ASTR + kk + khi + 1];
            // B frags (two 16-wide N tiles), mirrored layout
            v2f b0, b1;
            b0.x = Bt[(kk + khi)     * BSTR + ncol0];
            b0.y = Bt[(kk + khi + 1) * BSTR + ncol0];
            b1.x = Bt[(kk + khi)     * BSTR + ncol0 + 16];
            b1.y = Bt[(kk + khi + 1) * BSTR + ncol0 + 16];
            c0 = __builtin_amdgcn_wmma_f32_16x16x4_f32(false, a, false, b0,
                                                       (short)0, c0, false, false);
            c1 = __builtin_amdgcn_wmma_f32_16x16x4_f32(false, a, false, b1,
                                                       (short)0, c1, false, false);
        }
        __syncthreads();
    }

    // store: C VGPR v holds rows M=v (lanes 0-15) / M=v+8 (lanes 16-31)
    const int rhalf = (lane >> 4) * 8;
    const int ccol  = colBase + wn + (lane & 15);
    #pragma unroll
    for (int v = 0; v < 8; ++v) {
        long row = rowBase + wm + rhalf + v;
        C[row * NH + ccol]      = c0[v];
        C[row * NH + ccol + 16] = c1[v];
    }
}

// ---------------------------------------------------------------------------
// Per-row dual dot: s_src[r] = h[r]·a_src ; s_dst[r] = h[r]·a_dst
// one wave per row, 8 elements per lane
// ---------------------------------------------------------------------------
__global__ __launch_bounds__(256)
void dual_dot(const float* __restrict__ h, const float* __restrict__ a_src,
              const float* __restrict__ a_dst, float* __restrict__ ssrc,
              float* __restrict__ sdst, int nrows) {
    __shared__ float as[NH], ad[NH];
    if (threadIdx.x < NH) { as[threadIdx.x] = a_src[threadIdx.x];
                            ad[threadIdx.x] = a_dst[threadIdx.x]; }
    __syncthreads();
    int wave = threadIdx.x >> 5, lane = threadIdx.x & 31;
    int row = blockIdx.x * 8 + wave;
    if (row >= nrows) return;
    const float* hp = h + (size_t)row * NH;
    float s0 = 0.f, s1 = 0.f;
    #pragma unroll
    for (int j = 0; j < 8; ++j) {
        float v = hp[lane + 32 * j];
        s0 = fmaf(v, as[lane + 32 * j], s0);
        s1 = fmaf(v, ad[lane + 32 * j], s1);
    }
    #pragma unroll
    for (int o = 16; o; o >>= 1) {
        s0 += __shfl_xor(s0, o, 32);
        s1 += __shfl_xor(s1, o, 32);
    }
    if (lane == 0) { ssrc[row] = s0; sdst[row] = s1; }
}

// ---------------------------------------------------------------------------
// CSR build: zero, histogram, single-block scan, scatter
// ---------------------------------------------------------------------------
__global__ void zero_i32(int* __restrict__ p, int n) {
    int i = blockIdx.x * blockDim.x + threadIdx.x;
    if (i < n) p[i] = 0;
}

__global__ void count_edges(const int* __restrict__ tgt, int* __restrict__ counts, int nE) {
    int i = blockIdx.x * blockDim.x + threadIdx.x;
    if (i < nE) atomicAdd(&counts[tgt[i]], 1);
}

__global__ __launch_bounds__(1024)
void scan_offsets(const int* __restrict__ counts, int* __restrict__ offsets,
                  int* __restrict__ cursor, int n) {
    __shared__ int csum[1024];
    int t = threadIdx.x;
    int per = (n + 1023) >> 10;
    int b = t * per;
    int e = min(n, b + per);
    int s = 0;
    for (int i = b; i < e; ++i) s += counts[i];
    csum[t] = s;
    __syncthreads();
    if (t == 0) {
        int run = 0;
        for (int i = 0; i < 1024; ++i) { int v = csum[i]; csum[i] = run; run += v; }
        offsets[n] = run;
    }
    __syncthreads();
    int run = csum[t];
    for (int i = b; i < e; ++i) { offsets[i] = run; run += counts[i]; cursor[i] = 0; }
}

__global__ void scatter_edges(const int* __restrict__ tgt, const int* __restrict__ offsets,
                              int* __restrict__ cursor, int* __restrict__ sorted, int nE) {
    int i = blockIdx.x * blockDim.x + threadIdx.x;
    if (i < nE) {
        int t = tgt[i];
        int pos = offsets[t] + atomicAdd(&cursor[t], 1);
        sorted[pos] = i;
    }
}

// ---------------------------------------------------------------------------
// Fused per-target: leaky-relu scores -> softmax -> weighted row aggregation
// -> elu. One wave per target; 256-wide accumulator (8 floats/lane).
// ---------------------------------------------------------------------------
__global__ __launch_bounds__(256)
void agg_softmax_elu(const float* __restrict__ h, const float* __restrict__ ssrc,
                     const float* __restrict__ sdst, const int* __restrict__ eo,
                     const int* __restrict__ ei, const int* __restrict__ sorted,
                     const int* __restrict__ offsets, float* __restrict__ out, int ntgt) {
    int wave = threadIdx.x >> 5, lane = threadIdx.x & 31;
    int t = blockIdx.x * 8 + wave;
    if (t >= ntgt) return;
    int beg = offsets[t], end = offsets[t + 1];

    // pass 1: segment max of leaky-relu scores
    float m = -3.4e38f;
    for (int base = beg; base < end; base += 32) {
        int e = base + lane;
        float sc = -3.4e38f;
        if (e < end) {
            int id = sorted[e];
            float v = ssrc[eo[id]] + sdst[ei[id]];
            sc = v > 0.f ? v : ALPHA * v;
        }
        m = fmaxf(m, sc);
    }
    #pragma unroll
    for (int o = 16; o; o >>= 1) m = fmaxf(m, __shfl_xor(m, o, 32));

    // pass 2: exp, unnormalized aggregation of h rows
    float acc[8];
    #pragma unroll
    for (int j = 0; j < 8; ++j) acc[j] = 0.f;
    float ssum = 0.f;
    for (int base = beg; base < end; base += 32) {
        int e = base + lane;
        float ex = 0.f; int sidx = 0;
        if (e < end) {
            int id = sorted[e];
            float v = ssrc[eo[id]] + sdst[ei[id]];
            v = v > 0.f ? v : ALPHA * v;
            ex = __expf(v - m);
            sidx = ei[id];
        }
        ssum += ex;
        int cnt = min(32, end - base);
        for (int k = 0; k < cnt; ++k) {
            float w  = __shfl(ex, k, 32);
            int  src = __shfl(sidx, k, 32);
            const float* hp = h + (size_t)src * NH + lane;
            #pragma unroll
            for (int j = 0; j < 8; ++j) acc[j] = fmaf(w, hp[32 * j], acc[j]);
        }
    }
    #pragma unroll
    for (int o = 16; o; o >>= 1) ssum += __shfl_xor(ssum, o, 32);
    float inv = ssum > 0.f ? 1.f / ssum : 0.f;

    float* op = out + (size_t)t * NH + lane;
    #pragma unroll
    for (int j = 0; j < 8; ++j) {
        float v = acc[j] * inv;
        op[32 * j] = v > 0.f ? v : (__expf(v) - 1.f);   // elu
    }
}

// ---------------------------------------------------------------------------
// Jumping-knowledge max: out = max(xs1[tgt_samp1], xs2[tgt_samp0])
// ---------------------------------------------------------------------------
__global__ void jk_max(const float* __restrict__ xs1, const float* __restrict__ xs2,
                       const int* __restrict__ ts1, const int* __restrict__ ts0,
                       float* __restrict__ out) {
    int i = blockIdx.x * blockDim.x + threadIdx.x;   // N0S * NH total
    int row = i >> 8, col = i & 255;
    out[i] = fmaxf(xs1[(size_t)ts1[row] * NH + col],
                   xs2[(size_t)ts0[row] * NH + col]);
}

// ---------------------------------------------------------------------------
extern "C" void kernel_launch(void* const* d_in, const int* in_sizes, int n_in,
                              void* d_out, int out_size, void* d_ws, size_t ws_size,
                              hipStream_t stream) {
    const float* x          = (const float*)d_in[0];
    const int*   sample_last= (const int*)d_in[2];
    const int*   edge_out1  = (const int*)d_in[3];
    const int*   edge_in1   = (const int*)d_in[4];
    const int*   tgt_out1   = (const int*)d_in[5];
    const int*   tgt_samp1  = (const int*)d_in[7];
    const int*   edge_out0  = (const int*)d_in[8];
    const int*   edge_in0   = (const int*)d_in[9];
    const int*   tgt_out0   = (const int*)d_in[10];
    const int*   tgt_samp0  = (const int*)d_in[12];
    const float* W0         = (const float*)d_in[13];
    const float* a_src0     = (const float*)d_in[15];
    const float* a_dst0     = (const float*)d_in[16];
    const float* W1         = (const float*)d_in[17];
    const float* a_src1     = (const float*)d_in[19];
    const float* a_dst1     = (const float*)d_in[20];
    float* out = (float*)d_out;

    // workspace carve-up (256B aligned)
    char* ws = (char*)d_ws;
    size_t off = 0;
    auto alloc = [&](size_t bytes) {
        char* p = ws + off;
        off += (bytes + 255) & ~(size_t)255;
        return p;
    };
    float* h      = (float*)alloc((size_t)N2S * NH * sizeof(float));  // reused for h1
    float* xs1    = (float*)alloc((size_t)N1S * NH * sizeof(float));
    float* xs2    = (float*)alloc((size_t)N0S * NH * sizeof(float));
    float* ssrc   = (float*)alloc((size_t)N2S * sizeof(float));
    float* sdst   = (float*)alloc((size_t)N2S * sizeof(float));
    int*   counts = (int*)alloc((size_t)N1S * sizeof(int));
    int*   offs   = (int*)alloc((size_t)(N1S + 1) * sizeof(int));
    int*   cursor = (int*)alloc((size_t)N1S * sizeof(int));
    int*   sorted = (int*)alloc((size_t)E1S * sizeof(int));
    (void)ws_size; (void)in_sizes; (void)n_in; (void)out_size;

    // ---- layer 0: N2 sources -> N1 targets ----
    gat_gemm_wmma<<<dim3(N2S / 64, NH / 64), 256, 0, stream>>>(x, W0, sample_last, h);
    dual_dot<<<N2S / 8, 256, 0, stream>>>(h, a_src0, a_dst0, ssrc, sdst, N2S);
    zero_i32<<<(N1S + 255) / 256, 256, 0, stream>>>(counts, N1S);
    count_edges<<<(E1S + 255) / 256, 256, 0, stream>>>(tgt_out1, counts, E1S);
    scan_offsets<<<1, 1024, 0, stream>>>(counts, offs, cursor, N1S);
    scatter_edges<<<(E1S + 255) / 256, 256, 0, stream>>>(tgt_out1, offs, cursor, sorted, E1S);
    agg_softmax_elu<<<N1S / 8, 256, 0, stream>>>(h, ssrc, sdst, edge_out1, edge_in1,
                                                 sorted, offs, xs1, N1S);

    // ---- layer 1: N1 sources -> N0 targets ----
    gat_gemm_wmma<<<dim3(N1S / 64, NH / 64), 256, 0, stream>>>(xs1, W1, nullptr, h);
    dual_dot<<<N1S / 8, 256, 0, stream>>>(h, a_src1, a_dst1, ssrc, sdst, N1S);
    zero_i32<<<(N0S + 255) / 256, 256, 0, stream>>>(counts, N0S);
    count_edges<<<(E0S + 255) / 256, 256, 0, stream>>>(tgt_out0, counts, E0S);
    scan_offsets<<<1, 1024, 0, stream>>>(counts, offs, cursor, N0S);
    scatter_edges<<<(E0S + 255) / 256, 256, 0, stream>>>(tgt_out0, offs, cursor, sorted, E0S);
    agg_softmax_elu<<<N0S / 8, 256, 0, stream>>>(h, ssrc, sdst, edge_out0, edge_in0,
                                                 sorted, offs, xs2, N0S);

    // ---- jumping-knowledge max ----
    jk_max<<<(N0S * NH) / 256, 256, 0, stream>>>(xs1, xs2, tgt_samp1, tgt_samp0, out);
}